// IMF_90443421319621
// MI455X (gfx1250) — compile-verified
//
#include <hip/hip_runtime.h>
#include <hip/hip_bf16.h>
#include <math.h>

// ---------------------------------------------------------------------------
// IMF (quantized rank-32 factorization) for gfx1250 (MI455X), wave32 + WMMA.
// Batch = 32 matrices of 512x512 fp32.  All GEMMs use V_WMMA_F32_16X16X4_F32.
// ---------------------------------------------------------------------------

#define BATCH 32
#define MDIM  512
#define RANK  32
#define MATSZ (MDIM * MDIM)   // 262144
#define UVSZ  (MDIM * RANK)   // 16384

typedef float v2f __attribute__((ext_vector_type(2)));
typedef float v8f __attribute__((ext_vector_type(8)));

__device__ __forceinline__ v8f wmma4(v2f a, v2f b, v8f c) {
  // (neg_a, A, neg_b, B, c_mod, C, reuse_a, reuse_b)
  return __builtin_amdgcn_wmma_f32_16x16x4_f32(false, a, false, b, (short)0, c,
                                               false, false);
}

// ---------------------------------------------------------------------------
// Init V columns from rows of x (random Gaussian rows -> good subspace seed).
// ---------------------------------------------------------------------------
__global__ void k_init_v(const float* __restrict__ X, float* __restrict__ V) {
  size_t idx = (size_t)blockIdx.x * blockDim.x + threadIdx.x;  // B*512*32
  int b = (int)(idx >> 14);
  int e = (int)(idx & 16383);
  int n = e >> 5;
  int r = e & 31;
  V[(size_t)b * UVSZ + (size_t)n * RANK + r] =
      X[(size_t)b * MATSZ + (size_t)r * MDIM + n];
}

// ---------------------------------------------------------------------------
// C[512x32] = op(X)[512x512] @ Bm[512x32]   (WMMA f32 16x16x4)
// TRANS==0: A(m,k)=X[m*512+k] ; TRANS==1: A(m,k)=X[k*512+m]
// One wave per 16-row M-tile; both N-tiles computed off one A fragment.
// block = 256 (8 waves), grid = (4, BATCH).
// ---------------------------------------------------------------------------
template <int TRANS>
__global__ void k_gemm_xv(const float* __restrict__ Xall,
                          const float* __restrict__ Ball,
                          float* __restrict__ Call) {
  int b = blockIdx.y;
  const float* X = Xall + (size_t)b * MATSZ;
  const float* Bm = Ball + (size_t)b * UVSZ;
  float* C = Call + (size_t)b * UVSZ;

  int wave = threadIdx.x >> 5;
  int lane = threadIdx.x & 31;
  int m0 = (blockIdx.x * 8 + wave) * 16;   // 32 M-tiles
  int half = lane >> 4;
  int l16 = lane & 15;

  v8f c0 = {}, c1 = {};
  for (int k0 = 0; k0 < MDIM; k0 += 4) {
    int ka = k0 + 2 * half;
    v2f a;
    if (!TRANS) {
      a = *(const v2f*)(X + (size_t)(m0 + l16) * MDIM + ka);  // b64 load
    } else {
      a.x = X[(size_t)ka * MDIM + (m0 + l16)];
      a.y = X[(size_t)(ka + 1) * MDIM + (m0 + l16)];
    }
    v2f b0, b1;
    b0.x = Bm[(size_t)ka * RANK + l16];
    b0.y = Bm[(size_t)(ka + 1) * RANK + l16];
    b1.x = Bm[(size_t)ka * RANK + 16 + l16];
    b1.y = Bm[(size_t)(ka + 1) * RANK + 16 + l16];
    c0 = wmma4(a, b0, c0);                 // A fragment reused for both tiles
    c1 = wmma4(a, b1, c1);
  }
#pragma unroll
  for (int i = 0; i < 8; ++i) {
    int m = m0 + i + 8 * half;
    C[(size_t)m * RANK + l16] = c0[i];
    C[(size_t)m * RANK + 16 + l16] = c1[i];
  }
}

// ---------------------------------------------------------------------------
// G[32x32] = A^T A for A[512x32].  block = 128 (4 waves = 4 tiles), grid = B.
// ---------------------------------------------------------------------------
__global__ void k_gram(const float* __restrict__ Aall, float* __restrict__ Gall) {
  int b = blockIdx.x;
  const float* A = Aall + (size_t)b * UVSZ;
  float* G = Gall + (size_t)b * RANK * RANK;
  int wave = threadIdx.x >> 5;
  int lane = threadIdx.x & 31;
  int m0 = (wave >> 1) * 16;
  int n0 = (wave & 1) * 16;
  int half = lane >> 4;
  int l16 = lane & 15;

  v8f c = {};
  for (int k0 = 0; k0 < MDIM; k0 += 4) {
    int ka = k0 + 2 * half;
    v2f a, bb;
    a.x = A[(size_t)ka * RANK + m0 + l16];        // (A^T)(m,k) = A[k][m]
    a.y = A[(size_t)(ka + 1) * RANK + m0 + l16];
    bb.x = A[(size_t)ka * RANK + n0 + l16];
    bb.y = A[(size_t)(ka + 1) * RANK + n0 + l16];
    c = wmma4(a, bb, c);
  }
#pragma unroll
  for (int i = 0; i < 8; ++i)
    G[(size_t)(m0 + i + 8 * half) * RANK + n0 + l16] = c[i];
}

// ---------------------------------------------------------------------------
// In-place 32x32 Cholesky (lower) per batch; serial, tiny.
// ---------------------------------------------------------------------------
__global__ void k_chol(float* __restrict__ Gall) {
  if (threadIdx.x != 0) return;
  float* G = Gall + (size_t)blockIdx.x * RANK * RANK;
  for (int j = 0; j < RANK; ++j) {
    float d = G[j * RANK + j];
    for (int k = 0; k < j; ++k) d -= G[j * RANK + k] * G[j * RANK + k];
    d = sqrtf(fmaxf(d, 1e-20f));
    G[j * RANK + j] = d;
    float inv = 1.0f / d;
    for (int i = j + 1; i < RANK; ++i) {
      float s = G[i * RANK + j];
      for (int k = 0; k < j; ++k) s -= G[i * RANK + k] * G[j * RANK + k];
      G[i * RANK + j] = s * inv;
    }
  }
}

// ---------------------------------------------------------------------------
// V <- V * L^-T  (row-wise forward substitution).  grid=(2,B), block=256.
// ---------------------------------------------------------------------------
__global__ void k_trsm(float* __restrict__ Vall, const float* __restrict__ Lall) {
  int b = blockIdx.y;
  float* V = Vall + (size_t)b * UVSZ;
  const float* L = Lall + (size_t)b * RANK * RANK;
  __shared__ float Ls[RANK * RANK];
  for (int i = threadIdx.x; i < RANK * RANK; i += blockDim.x) Ls[i] = L[i];
  __syncthreads();
  int m = blockIdx.x * blockDim.x + threadIdx.x;  // 0..511
  float y[RANK];
#pragma unroll
  for (int j = 0; j < RANK; ++j) {
    float s = V[(size_t)m * RANK + j];
#pragma unroll
    for (int k = 0; k < RANK; ++k)
      if (k < j) s -= y[k] * Ls[j * RANK + k];
    y[j] = s / Ls[j * RANK + j];
  }
#pragma unroll
  for (int j = 0; j < RANK; ++j) V[(size_t)m * RANK + j] = y[j];
}

// ---------------------------------------------------------------------------
// Column L2 norms of U -> cs[b*32+r]  (approx singular values). block=32.
// ---------------------------------------------------------------------------
__global__ void k_colnorm(const float* __restrict__ Uall, float* __restrict__ cs) {
  int b = blockIdx.x, r = threadIdx.x;
  const float* U = Uall + (size_t)b * UVSZ;
  float s = 0.f;
  for (int m = 0; m < MDIM; ++m) {
    float t = U[(size_t)m * RANK + r];
    s += t * t;
  }
  cs[b * RANK + r] = sqrtf(s);
}

// u = U/sqrt(sigma), v = V*sqrt(sigma)
__global__ void k_uv_form(float* __restrict__ U, float* __restrict__ V,
                          const float* __restrict__ norms) {
  size_t idx = (size_t)blockIdx.x * blockDim.x + threadIdx.x;
  int b = (int)(idx >> 14);
  int e = (int)(idx & 16383);
  int r = e & 31;
  float sq = sqrtf(norms[b * RANK + r] + 1e-12f);
  U[(size_t)b * UVSZ + e] /= sq;
  V[(size_t)b * UVSZ + e] *= sq;
}

// per-batch min/max of u and v -> scales[b*2] = scale_u, [b*2+1] = scale_v
__global__ void k_minmax(const float* __restrict__ Uall,
                         const float* __restrict__ Vall,
                         float* __restrict__ scales) {
  int b = blockIdx.x;
  const float* U = Uall + (size_t)b * UVSZ;
  const float* V = Vall + (size_t)b * UVSZ;
  float umin = 3.4e38f, umax = -3.4e38f, vmin = 3.4e38f, vmax = -3.4e38f;
  for (int i = threadIdx.x; i < UVSZ; i += blockDim.x) {
    float u = U[i], v = V[i];
    umin = fminf(umin, u); umax = fmaxf(umax, u);
    vmin = fminf(vmin, v); vmax = fmaxf(vmax, v);
  }
  __shared__ float red[256];
  float vals[4] = {umin, -umax, vmin, -vmax};  // reduce all as min
#pragma unroll
  for (int p = 0; p < 4; ++p) {
    red[threadIdx.x] = vals[p];
    __syncthreads();
    for (int s = 128; s > 0; s >>= 1) {
      if ((int)threadIdx.x < s)
        red[threadIdx.x] = fminf(red[threadIdx.x], red[threadIdx.x + s]);
      __syncthreads();
    }
    vals[p] = red[0];
    __syncthreads();
  }
  if (threadIdx.x == 0) {
    float su = (-vals[1] - vals[0]) / 256.0f;
    float sv = (-vals[3] - vals[2]) / 256.0f;
    scales[b * 2 + 0] = fmaxf(su, 1e-30f);
    scales[b * 2 + 1] = fmaxf(sv, 1e-30f);
  }
}

__global__ void k_apply_scale(float* __restrict__ U, float* __restrict__ V,
                              const float* __restrict__ scales,
                              float* __restrict__ w) {
  size_t idx = (size_t)blockIdx.x * blockDim.x + threadIdx.x;
  int b = (int)(idx >> 14);
  int e = (int)(idx & 16383);
  float su = scales[b * 2], sv = scales[b * 2 + 1];
  U[(size_t)b * UVSZ + e] /= su;
  V[(size_t)b * UVSZ + e] /= sv;
  if (e == 0) {
    w[b * 2 + 0] = 0.0f;
    w[b * 2 + 1] = su * sv;
  }
}

// column sums of a 512x32 matrix. block=32, grid=B.
__global__ void k_colsum(const float* __restrict__ Mall, float* __restrict__ cs) {
  int b = blockIdx.x, r = threadIdx.x;
  const float* A = Mall + (size_t)b * UVSZ;
  float s = 0.f;
  for (int m = 0; m < MDIM; ++m) s += A[(size_t)m * RANK + r];
  cs[b * RANK + r] = s;
}

// ---------------------------------------------------------------------------
// Coordinate descent (l1=l2=0): row-local rank loop, one thread per row.
// a_eff[m][r] = (Araw[m][r] - w0*colsum[r]) / w1 ; sequential over r:
//   t2 = dot(urow, b[:,r]) - urow[r]*brr ; urow[r] = clip(round((a-t2+e)/(brr+e)))
// grid=(2,B), block=256.
// ---------------------------------------------------------------------------
__global__ void k_cd(float* __restrict__ Uall, const float* __restrict__ Aall,
                     const float* __restrict__ Ball,
                     const float* __restrict__ csall,
                     const float* __restrict__ wall) {
  int b = blockIdx.y;
  float* U = Uall + (size_t)b * UVSZ;
  const float* A = Aall + (size_t)b * UVSZ;
  const float* Bm = Ball + (size_t)b * RANK * RANK;
  const float* cs = csall + b * RANK;
  __shared__ float Bs[RANK * RANK];
  for (int i = threadIdx.x; i < RANK * RANK; i += blockDim.x) Bs[i] = Bm[i];
  __syncthreads();

  float w0 = wall[b * 2 + 0];
  float w1 = wall[b * 2 + 1];
  if (fabsf(w1) < 1e-16f) w1 = 1e-16f;
  float inv_w1 = 1.0f / w1;

  int m = blockIdx.x * blockDim.x + threadIdx.x;  // 0..511
  float urow[RANK], arow[RANK];
#pragma unroll
  for (int r = 0; r < RANK; ++r) {
    urow[r] = U[(size_t)m * RANK + r];
    arow[r] = (A[(size_t)m * RANK + r] - w0 * cs[r]) * inv_w1;
  }
#pragma unroll
  for (int r = 0; r < RANK; ++r) {
    float brr = Bs[r * RANK + r];
    float t2 = 0.f;
#pragma unroll
    for (int k = 0; k < RANK; ++k) t2 += urow[k] * Bs[k * RANK + r];
    t2 -= urow[r] * brr;
    float num = arow[r] - t2;                  // soft_threshold with l1=0
    float ur = (num + 1e-16f) / (brr + 1e-16f);
    urow[r] = fminf(fmaxf(rintf(ur), -128.0f), 127.0f);
  }
#pragma unroll
  for (int r = 0; r < RANK; ++r) U[(size_t)m * RANK + r] = urow[r];
}

// ---------------------------------------------------------------------------
// stats: [b*4 + {0:sz, 1:szz, 2:szx, 3:sx}] as double.
// ---------------------------------------------------------------------------
__global__ void k_zero_stats(double* __restrict__ stats) {
  int t = threadIdx.x;
  if (t < BATCH * 3) stats[(t / 3) * 4 + (t % 3)] = 0.0;
}

__global__ void k_sumx(const float* __restrict__ Xall, double* __restrict__ stats) {
  int b = blockIdx.x;
  const float* X = Xall + (size_t)b * MATSZ;
  double p = 0.0;
  for (int i = threadIdx.x; i < MATSZ; i += blockDim.x) p += (double)X[i];
  __shared__ double red[256];
  red[threadIdx.x] = p;
  __syncthreads();
  for (int s = 128; s > 0; s >>= 1) {
    if ((int)threadIdx.x < s) red[threadIdx.x] += red[threadIdx.x + s];
    __syncthreads();
  }
  if (threadIdx.x == 0) stats[b * 4 + 3] = red[0];
}

// ---------------------------------------------------------------------------
// z = u v^T via WMMA with A-fragment reuse: each wave owns one 16-row M-tile,
// preloads all 8 A fragments (K=32) as b64, sweeps all 32 N-tiles.
// Accumulate sz/szz/szx in f64; atomic f64 add to stats.
// grid=(4,B), block=256 (8 waves); 32 waves <-> 32 M-tiles.
// ---------------------------------------------------------------------------
__global__ void k_zstats(const float* __restrict__ Uall,
                         const float* __restrict__ Vall,
                         const float* __restrict__ Xall,
                         double* __restrict__ stats) {
  int b = blockIdx.y;
  const float* U = Uall + (size_t)b * UVSZ;
  const float* V = Vall + (size_t)b * UVSZ;
  const float* X = Xall + (size_t)b * MATSZ;
  int wave = threadIdx.x >> 5;
  int lane = threadIdx.x & 31;
  int half = lane >> 4, l16 = lane & 15;
  int m0 = (blockIdx.x * 8 + wave) * 16;  // M-tile

  v2f afr[8];
#pragma unroll
  for (int kk = 0; kk < 8; ++kk)
    afr[kk] = *(const v2f*)(U + (size_t)(m0 + l16) * RANK + kk * 4 + 2 * half);

  double psz = 0.0, pszz = 0.0, pszx = 0.0;
  for (int nt = 0; nt < 32; ++nt) {
    int n0 = nt * 16;
    v8f c = {};
#pragma unroll
    for (int kk = 0; kk < 8; ++kk) {
      v2f bb = *(const v2f*)(V + (size_t)(n0 + l16) * RANK + kk * 4 + 2 * half);
      c = wmma4(afr[kk], bb, c);
    }
#pragma unroll
    for (int i = 0; i < 8; ++i) {
      int m = m0 + i + 8 * half, n = n0 + l16;
      float z = c[i];
      float xv = X[(size_t)m * MDIM + n];
      psz += (double)z;
      pszz += (double)z * (double)z;
      pszx += (double)z * (double)xv;
    }
  }
  __shared__ double red[256];
  double vals[3] = {psz, pszz, pszx};
#pragma unroll
  for (int p = 0; p < 3; ++p) {
    red[threadIdx.x] = vals[p];
    __syncthreads();
    for (int s = 128; s > 0; s >>= 1) {
      if ((int)threadIdx.x < s) red[threadIdx.x] += red[threadIdx.x + s];
      __syncthreads();
    }
    vals[p] = red[0];
    __syncthreads();
  }
  if (threadIdx.x == 0) {
    atomicAdd(&stats[b * 4 + 0], vals[0]);
    atomicAdd(&stats[b * 4 + 1], vals[1]);
    atomicAdd(&stats[b * 4 + 2], vals[2]);
  }
}

__global__ void k_wsolve(const double* __restrict__ stats, float* __restrict__ w) {
  int b = threadIdx.x;
  if (b >= BATCH) return;
  double sz = stats[b * 4 + 0], szz = stats[b * 4 + 1];
  double szx = stats[b * 4 + 2], sx = stats[b * 4 + 3];
  double n = (double)MATSZ;
  double det = n * szz - sz * sz;
  if (fabs(det) < 1e-16) det = 1e-16;
  w[b * 2 + 1] = (float)((n * szx - sz * sx) / det);
  w[b * 2 + 0] = (float)((szz * sx - sz * szx) / det);
}

// out = w0 + w1 * (u v^T); same A-fragment-reuse tiling as k_zstats.
__global__ void k_final(const float* __restrict__ Uall,
                        const float* __restrict__ Vall,
                        const float* __restrict__ wall,
                        float* __restrict__ out) {
  int b = blockIdx.y;
  const float* U = Uall + (size_t)b * UVSZ;
  const float* V = Vall + (size_t)b * UVSZ;
  float* O = out + (size_t)b * MATSZ;
  float w0 = wall[b * 2 + 0], w1 = wall[b * 2 + 1];
  int wave = threadIdx.x >> 5;
  int lane = threadIdx.x & 31;
  int half = lane >> 4, l16 = lane & 15;
  int m0 = (blockIdx.x * 8 + wave) * 16;

  v2f afr[8];
#pragma unroll
  for (int kk = 0; kk < 8; ++kk)
    afr[kk] = *(const v2f*)(U + (size_t)(m0 + l16) * RANK + kk * 4 + 2 * half);

  for (int nt = 0; nt < 32; ++nt) {
    int n0 = nt * 16;
    v8f c = {};
#pragma unroll
    for (int kk = 0; kk < 8; ++kk) {
      v2f bb = *(const v2f*)(V + (size_t)(n0 + l16) * RANK + kk * 4 + 2 * half);
      c = wmma4(afr[kk], bb, c);
    }
#pragma unroll
    for (int i = 0; i < 8; ++i) {
      int m = m0 + i + 8 * half, n = n0 + l16;
      O[(size_t)m * MDIM + n] = w0 + w1 * c[i];
    }
  }
}

// ---------------------------------------------------------------------------
// Workspace layout (floats; stats region is 8B-aligned doubles)
// ---------------------------------------------------------------------------
static constexpr size_t OFF_U  = 0;
static constexpr size_t OFF_V  = OFF_U + (size_t)BATCH * UVSZ;
static constexpr size_t OFF_A  = OFF_V + (size_t)BATCH * UVSZ;
static constexpr size_t OFF_B  = OFF_A + (size_t)BATCH * UVSZ;
static constexpr size_t OFF_CS = OFF_B + (size_t)BATCH * RANK * RANK;
static constexpr size_t OFF_W  = OFF_CS + (size_t)BATCH * RANK;
static constexpr size_t OFF_SC = OFF_W + (size_t)BATCH * 2;
static constexpr size_t OFF_ST = (OFF_SC + (size_t)BATCH * 2 + 1) & ~(size_t)1;

extern "C" void kernel_launch(void* const* d_in, const int* in_sizes, int n_in,
                              void* d_out, int out_size, void* d_ws,
                              size_t ws_size, hipStream_t stream) {
  (void)in_sizes; (void)n_in; (void)out_size; (void)ws_size;
  const float* X = (const float*)d_in[0];
  float* out = (float*)d_out;
  float* ws = (float*)d_ws;
  float* U  = ws + OFF_U;
  float* V  = ws + OFF_V;
  float* A  = ws + OFF_A;
  float* Bm = ws + OFF_B;
  float* CS = ws + OFF_CS;
  float* W  = ws + OFF_W;
  float* SC = ws + OFF_SC;
  double* ST = (double*)(ws + OFF_ST);

  dim3 blk256(256);
  dim3 gElem(2048);            // B*512*32 / 256
  dim3 gGemm(4, BATCH);
  dim3 gRows(2, BATCH);
  dim3 gTile(4, BATCH);

  // ---- init: subspace iteration (SVD surrogate) ----
  k_init_v<<<gElem, blk256, 0, stream>>>(X, V);
  for (int it = 0; it < 2; ++it) {
    k_gemm_xv<0><<<gGemm, blk256, 0, stream>>>(X, V, A);   // T = X V
    k_gemm_xv<1><<<gGemm, blk256, 0, stream>>>(X, A, V);   // V = X^T T
    k_gram<<<BATCH, 128, 0, stream>>>(V, Bm);
    k_chol<<<BATCH, 32, 0, stream>>>(Bm);
    k_trsm<<<gRows, blk256, 0, stream>>>(V, Bm);
  }
  k_gemm_xv<0><<<gGemm, blk256, 0, stream>>>(X, V, U);     // U = X V
  k_colnorm<<<BATCH, 32, 0, stream>>>(U, CS);              // sigma
  k_uv_form<<<gElem, blk256, 0, stream>>>(U, V, CS);
  k_minmax<<<BATCH, 256, 0, stream>>>(U, V, SC);
  k_apply_scale<<<gElem, blk256, 0, stream>>>(U, V, SC, W);
  k_sumx<<<BATCH, 256, 0, stream>>>(X, ST);

  // ---- 10 alternating CD + w refit iterations ----
  for (int it = 0; it < 10; ++it) {
    k_gemm_xv<0><<<gGemm, blk256, 0, stream>>>(X, V, A);   // a_raw = X v
    k_gram<<<BATCH, 128, 0, stream>>>(V, Bm);              // b = v^T v
    k_colsum<<<BATCH, 32, 0, stream>>>(V, CS);
    k_cd<<<gRows, blk256, 0, stream>>>(U, A, Bm, CS, W);

    k_gemm_xv<1><<<gGemm, blk256, 0, stream>>>(X, U, A);   // a_raw = X^T u
    k_gram<<<BATCH, 128, 0, stream>>>(U, Bm);              // b = u^T u
    k_colsum<<<BATCH, 32, 0, stream>>>(U, CS);
    k_cd<<<gRows, blk256, 0, stream>>>(V, A, Bm, CS, W);

    k_zero_stats<<<1, 128, 0, stream>>>(ST);
    k_zstats<<<gTile, blk256, 0, stream>>>(U, V, X, ST);
    k_wsolve<<<1, 32, 0, stream>>>(ST, W);
  }

  // ---- output ----
  k_final<<<gTile, blk256, 0, stream>>>(U, V, W, out);
}